// Postprocessing_landmark_torch_53961969107561
// MI455X (gfx1250) — compile-verified
//
#include <hip/hip_runtime.h>
#include <stdint.h>

#define TD 1024
#define RB 16                    // output rows per block
#define LDSW 1028                // padded LDS row stride (floats): 1028*4 % 16 == 0
#define K 10
#define NTHREADS 256
#define KP_C 7
#define LM_C 4
#define KP_SLOTS ((TD / RB) * KP_C)   // 448 partial lists
#define LM_SLOTS ((TD / RB) * LM_C)   // 256 partial lists

struct Pair { float v; int i; };
static_assert(sizeof(Pair) == 8, "Pair must be 8 bytes");

// ---------------- CDNA5 async global->LDS copy helpers ----------------
#if defined(__has_builtin)
#if __has_builtin(__builtin_amdgcn_global_load_async_to_lds_b128)
#define HAVE_ASYNC_BUILTIN 1
#endif
#if __has_builtin(__builtin_amdgcn_s_wait_asynccnt)
#define HAVE_WAIT_ASYNC_BUILTIN 1
#endif
#endif

typedef int v4i __attribute__((vector_size(16)));                 // 4 x i32 (b128 payload)
typedef v4i __attribute__((address_space(1))) gv4i;               // global
typedef v4i __attribute__((address_space(3))) lv4i;               // LDS

__device__ __forceinline__ void async_g2l_b128(const float* g, float* l) {
#ifdef HAVE_ASYNC_BUILTIN
  __builtin_amdgcn_global_load_async_to_lds_b128(
      (gv4i*)(uintptr_t)g,
      (lv4i*)(uint32_t)(uintptr_t)l,   // low 32 bits of generic = LDS byte offset
      0, 0);
#else
  uint32_t lds_addr = (uint32_t)(uintptr_t)l;
  asm volatile("global_load_async_to_lds_b128 %0, %1, off"
               :: "v"(lds_addr), "v"(g) : "memory");
#endif
}

__device__ __forceinline__ void wait_async0() {
#ifdef HAVE_WAIT_ASYNC_BUILTIN
  __builtin_amdgcn_s_wait_asynccnt(0);
#else
  asm volatile("s_wait_asynccnt 0" ::: "memory");
#endif
}

// ---------------- top-k primitives (value desc, index asc — matches jax top_k) ----
__device__ __forceinline__ bool pbetter(float v1, int i1, float v2, int i2) {
  return (v1 > v2) || ((v1 == v2) && (i1 < i2));
}

__device__ __forceinline__ void topk_insert(float* lv, int* li, float& vmin, int& imin,
                                            float v, int i) {
  if (!pbetter(v, i, vmin, imin)) return;     // fast reject against register copy of 10th
  int p = K - 1;
  while (p > 0 && pbetter(v, i, lv[p - 1], li[p - 1])) {
    lv[p] = lv[p - 1]; li[p] = li[p - 1]; --p;
  }
  lv[p] = v; li[p] = i;
  vmin = lv[K - 1]; imin = li[K - 1];
}

__device__ __forceinline__ void topk_init(float* lv, int* li, float& vmin, int& imin) {
  const float NEG = -__builtin_inff();
#pragma unroll
  for (int k = 0; k < K; ++k) { lv[k] = NEG; li[k] = 0x7fffffff; }
  vmin = NEG; imin = 0x7fffffff;
}

// ---------------- kernel 1: peak suppression + per-block partial top-10 ----------
__global__ __launch_bounds__(NTHREADS) void peak_topk_partial(
    const float* __restrict__ map, int C, Pair* __restrict__ out) {
  __shared__ __align__(16) float tile[(RB + 2) * LDSW];   // 74,016 B
  const int   tid = threadIdx.x;
  const int   c   = blockIdx.y;
  const int   yb  = blockIdx.x;
  const float NEG = -__builtin_inff();
  const float* plane = map + (size_t)c * TD * TD;

  // Async-load RB+2 rows (halo of 1); out-of-range rows filled with -inf.
#pragma unroll 1
  for (int i = 0; i < RB + 2; ++i) {
    int gy = yb * RB - 1 + i;
    float* dst = &tile[i * LDSW + 4 * tid];               // 16B-aligned LDS dest
    if (gy >= 0 && gy < TD) {
      async_g2l_b128(plane + (size_t)gy * TD + 4 * tid, dst);
    } else {
      *(float4*)dst = make_float4(NEG, NEG, NEG, NEG);
    }
  }
  if (tid < RB + 2) {                                     // -inf halo columns
    tile[tid * LDSW + 1024] = NEG;
    tile[tid * LDSW + 1025] = NEG;
  }
  wait_async0();
  __syncthreads();

  float lv[K]; int li[K]; float vmin; int imin;
  topk_init(lv, li, vmin, imin);

  // Rolling vertical 3x3 max: 4 columns per thread, 16 rows each.
#pragma unroll 1
  for (int j = 0; j < 4; ++j) {
    int x  = tid + 256 * j;
    int cC = x;
    int cL = (x == 0)    ? 1024 : x - 1;                  // -inf halo cells
    int cR = (x == 1023) ? 1025 : x + 1;
    float hm0 = fmaxf(fmaxf(tile[0 * LDSW + cL], tile[0 * LDSW + cC]), tile[0 * LDSW + cR]);
    float hm1 = fmaxf(fmaxf(tile[1 * LDSW + cL], tile[1 * LDSW + cC]), tile[1 * LDSW + cR]);
#pragma unroll 4
    for (int r = 0; r < RB; ++r) {
      const float* rp = &tile[(r + 2) * LDSW];
      float hm2 = fmaxf(fmaxf(rp[cL], rp[cC]), rp[cR]);
      float v   = tile[(r + 1) * LDSW + cC];
      float m   = fmaxf(hm0, fmaxf(hm1, hm2));            // m >= v always
      float val = ((m - v) < 1e-6f) ? v : 0.0f;           // peak keep, else 0
      int   flat = ((yb * RB + r) * TD + x) * C + c;      // HWC flatten index
      topk_insert(lv, li, vmin, imin, val, flat);
      hm0 = hm1; hm1 = hm2;
    }
  }

  // Block reduce 256 -> 32 -> 1 lists (reuse tile as Pair scratch).
  __syncthreads();
  Pair* buf  = (Pair*)tile;                // 256*10 pairs
  Pair* buf2 = buf + NTHREADS * K;         // 32*10 pairs  (total 23,040 B < 74,016 B)
#pragma unroll
  for (int k = 0; k < K; ++k) buf[tid * K + k] = {lv[k], li[k]};
  __syncthreads();

  if (tid < 32) {
    float mv[K]; int mi[K]; float mmin; int mimin;
    topk_init(mv, mi, mmin, mimin);
#pragma unroll 1
    for (int s = 0; s < 8; ++s)
#pragma unroll 1
      for (int k = 0; k < K; ++k) {
        Pair p = buf[(tid * 8 + s) * K + k];
        topk_insert(mv, mi, mmin, mimin, p.v, p.i);
      }
    for (int k = 0; k < K; ++k) buf2[tid * K + k] = {mv[k], mi[k]};
  }
  __syncthreads();

  if (tid == 0) {
    float fv[K]; int fi[K]; float fmin; int fimin;
    topk_init(fv, fi, fmin, fimin);
#pragma unroll 1
    for (int s = 0; s < 32; ++s)
#pragma unroll 1
      for (int k = 0; k < K; ++k) {
        Pair p = buf2[s * K + k];
        topk_insert(fv, fi, fmin, fimin, p.v, p.i);
      }
    int slot = blockIdx.y * gridDim.x + blockIdx.x;
    for (int k = 0; k < K; ++k) out[slot * K + k] = {fv[k], fi[k]};
  }
}

// ---------------- kernel 2: final top-10 merge + decode ----------------
__global__ __launch_bounds__(NTHREADS) void finalize_kernel(
    const Pair* __restrict__ ws,
    const float* __restrict__ offset,
    const float* __restrict__ size_,
    const float* __restrict__ lmoff,
    float* __restrict__ out) {
  __shared__ Pair buf[NTHREADS * K + 32 * K];
  __shared__ Pair fin[2][K];
  const int tid = threadIdx.x;

#pragma unroll 1
  for (int ph = 0; ph < 2; ++ph) {
    const Pair* cand = ws + (ph == 0 ? 0 : KP_SLOTS * K);
    const int   n    = (ph == 0 ? KP_SLOTS : LM_SLOTS) * K;

    float lv[K]; int li[K]; float vmin; int imin;
    topk_init(lv, li, vmin, imin);
#pragma unroll 1
    for (int e = tid; e < n; e += NTHREADS) {
      Pair p = cand[e];
      topk_insert(lv, li, vmin, imin, p.v, p.i);
    }
#pragma unroll
    for (int k = 0; k < K; ++k) buf[tid * K + k] = {lv[k], li[k]};
    __syncthreads();

    if (tid < 32) {
      float mv[K]; int mi[K]; float mmin; int mimin;
      topk_init(mv, mi, mmin, mimin);
#pragma unroll 1
      for (int s = 0; s < 8; ++s)
#pragma unroll 1
        for (int k = 0; k < K; ++k) {
          Pair p = buf[(tid * 8 + s) * K + k];
          topk_insert(mv, mi, mmin, mimin, p.v, p.i);
        }
      for (int k = 0; k < K; ++k) buf[NTHREADS * K + tid * K + k] = {mv[k], mi[k]};
    }
    __syncthreads();

    if (tid == 0) {
      float fv[K]; int fi[K]; float fmin; int fimin;
      topk_init(fv, fi, fmin, fimin);
#pragma unroll 1
      for (int s = 0; s < 32; ++s)
#pragma unroll 1
        for (int k = 0; k < K; ++k) {
          Pair p = buf[NTHREADS * K + s * K + k];
          topk_insert(fv, fi, fmin, fimin, p.v, p.i);
        }
      for (int k = 0; k < K; ++k) fin[ph][k] = {fv[k], fi[k]};
    }
    __syncthreads();
  }

  // ---- decode & write output (tuple order, 100 floats total) ----
  if (tid < K) {
    // detection branch
    Pair p = fin[0][tid];
    int idx = p.i;
    int t   = idx / KP_C;
    int cls = idx - t * KP_C;
    int y   = t >> 10;
    int x   = t & (TD - 1);
    size_t base = (size_t)y * TD + x;
    float off0 = offset[base];                       // x-offset (channel 0)
    float off1 = offset[(size_t)TD * TD + base];     // y-offset (channel 1)
    float s0   = size_[base];
    float s1   = size_[(size_t)TD * TD + base];
    float posy = (float)y + off1;
    float posx = (float)x + off0;
    float hy = fmaxf(s1, 0.0f) * 0.5f;
    float hx = fmaxf(s0, 0.0f) * 0.5f;
    float ymin = fminf(fmaxf(posy - hy, 0.0f), (float)(TD - 1)) * 4.0f;
    float xmin = fminf(fmaxf(posx - hx, 0.0f), (float)(TD - 1)) * 4.0f;
    float ymax = fminf(fmaxf(posy + hy, 0.0f), (float)(TD - 1)) * 4.0f;
    float xmax = fminf(fmaxf(posx + hx, 0.0f), (float)(TD - 1)) * 4.0f;
    out[tid * 4 + 0] = ymin;
    out[tid * 4 + 1] = xmin;
    out[tid * 4 + 2] = ymax;
    out[tid * 4 + 3] = xmax;
    out[40 + tid] = (float)cls;   // det_classes
    out[50 + tid] = p.v;          // det_scores
  } else if (tid < 2 * K) {
    // landmark branch
    int  u   = tid - K;
    Pair p   = fin[1][u];
    int idx  = p.i;
    int t    = idx >> 2;          // / N_LANDMARK
    int cls  = idx & 3;
    int y    = t >> 10;
    int x    = t & (TD - 1);
    size_t base = (size_t)y * TD + x;
    float lo0 = lmoff[base];
    float lo1 = lmoff[(size_t)TD * TD + base];
    out[60 + 2 * u]     = ((float)x + lo0) * 4.0f;  // lm_points (x, y)
    out[60 + 2 * u + 1] = ((float)y + lo1) * 4.0f;
    out[80 + u] = (float)cls;     // lm_classes
    out[90 + u] = p.v;            // lm_conf
  }
}

extern "C" void kernel_launch(void* const* d_in, const int* in_sizes, int n_in,
                              void* d_out, int out_size, void* d_ws, size_t ws_size,
                              hipStream_t stream) {
  const float* offset   = (const float*)d_in[0];
  const float* size_    = (const float*)d_in[1];
  const float* keypoint = (const float*)d_in[2];
  const float* landmark = (const float*)d_in[3];
  const float* lmoff    = (const float*)d_in[4];
  Pair* ws = (Pair*)d_ws;   // needs (448+256)*10*8 = 56,320 bytes

  peak_topk_partial<<<dim3(TD / RB, KP_C), NTHREADS, 0, stream>>>(keypoint, KP_C, ws);
  peak_topk_partial<<<dim3(TD / RB, LM_C), NTHREADS, 0, stream>>>(landmark, LM_C,
                                                                  ws + KP_SLOTS * K);
  finalize_kernel<<<1, NTHREADS, 0, stream>>>(ws, offset, size_, lmoff, (float*)d_out);
}